// DETRLoss_56908316672027
// MI455X (gfx1250) — compile-verified
//
#include <hip/hip_runtime.h>
#include <hip/hip_bf16.h>
#include <math.h>

typedef float v2f __attribute__((ext_vector_type(2)));
typedef float v4f __attribute__((ext_vector_type(4)));
typedef float v8f __attribute__((ext_vector_type(8)));

#define NB 8
#define NQ 300
#define CCH 92
#define CF 2048
#define HF 42
#define MM 20
#define NTOPK 5
#define NCLS 91
#define HW 1333
#define PIX (HF*HF)            /* 1764 = 441 float4 */
#define NEGV (-1.0e11f)
#define MPAD 304               /* 300 rows padded to 19*16 */
#define KPAD 44                /* 42 padded to 11*4 */
#define NPAD 48                /* 42 padded to 3*16 */

/* workspace layout (float offsets) */
#define OFF_CHSUM  0                            /* 8*1764            */
#define OFF_WTAB   (OFF_CHSUM + NB*PIX)         /* 1334*42           */
#define OFF_DY     (OFF_WTAB + (HW+1)*HF)       /* 8*304*44          */
#define OFF_DX     (OFF_DY + NB*MPAD*KPAD)      /* 8*300*42          */
#define OFF_FP     (OFF_DX + NB*NQ*HF)          /* 8*44*48           */
#define OFF_U      (OFF_FP + NB*KPAD*NPAD)      /* 8*304*48          */
#define OFF_MEANS  (OFF_U + NB*MPAD*NPAD)       /* 2400              */
#define OFF_CNT    (OFF_MEANS + NB*NQ)          /* 2400              */

/* ---------------- init: zero channel-sum accumulator + output -------- */
__global__ void k_init(float* __restrict__ ws, float* __restrict__ out) {
  int idx = blockIdx.x * blockDim.x + threadIdx.x;
  if (idx < NB * PIX) ws[OFF_CHSUM + idx] = 0.f;
  if (idx == 0) out[0] = 0.f;
}

/* ------- cumulative bilinear weight table W[1334][42] (half-pixel) --- */
__global__ void k_weights(float* __restrict__ ws) {
  int s = threadIdx.x;
  if (s >= HF) return;
  float* WT = ws + OFF_WTAB;
  const float r = (float)HF / (float)HW;
  float run = 0.f;
  WT[s] = 0.f;
  for (int y = 1; y <= HW; ++y) {
    float x = ((float)(y - 1) + 0.5f) * r - 0.5f;
    float fl = floorf(x);
    int s0 = (int)fl;
    float f = x - fl;
    int s0c = min(max(s0, 0), HF - 1);
    int s1c = min(max(s0 + 1, 0), HF - 1);
    float wgt = 0.f;
    if (s == s0c) wgt += 1.f - f;
    if (s == s1c) wgt += f;
    run += wgt;
    WT[y * HF + s] = run;
  }
}

/* ---- bandwidth-dominant phase: channel sum of img_features ----------
 * grid (16 chunks of 128 ch, 8 batches) x 256 threads.
 * v4f streaming: 512B coalesced per wave-issue (global_load_b128),
 * non-temporal hint since each byte is touched exactly once.           */
__global__ void k_chansum(const float* __restrict__ img, float* __restrict__ ws) {
  float* CH = ws + OFF_CHSUM;
  int b = blockIdx.y;
  int chunk = blockIdx.x;                 /* 0..15, 128 channels each */
  int t = threadIdx.x;                    /* 256 */
  const v4f* base =
      (const v4f*)(img + ((size_t)b * CF + (size_t)chunk * 128) * PIX);
  const int NV4 = PIX / 4;                /* 441 */
  #pragma unroll
  for (int s = 0; s < 2; ++s) {
    int v4 = t + s * 256;
    if (v4 < NV4) {
      v4f a = {0.f, 0.f, 0.f, 0.f};
      #pragma unroll 4
      for (int c = 0; c < 128; ++c) {
        v4f x = __builtin_nontemporal_load(&base[(size_t)c * NV4 + v4]);
        a += x;
      }
      float* dst = &CH[b * PIX + v4 * 4];
      atomicAdd(dst + 0, a.x);
      atomicAdd(dst + 1, a.y);
      atomicAdd(dst + 2, a.z);
      atomicAdd(dst + 3, a.w);
    }
  }
}

/* ---- pad channel-sum into F (44x48 per batch, zero padded) ---------- */
__global__ void k_fpad(float* __restrict__ ws) {
  const float* CH = ws + OFF_CHSUM;
  float* FP = ws + OFF_FP;
  int b = blockIdx.x;
  for (int idx = threadIdx.x; idx < KPAD * NPAD; idx += blockDim.x) {
    int k = idx / NPAD, n = idx % NPAD;
    FP[(size_t)b * KPAD * NPAD + idx] =
        (k < HF && n < HF) ? CH[b * PIX + k * HF + n] : 0.f;
  }
}

/* ---- per-box cumulative-weight difference vectors Dy, Dx ------------ */
__global__ void k_boxvec(const float* __restrict__ pb, const int* __restrict__ hp,
                         const int* __restrict__ wp, float* __restrict__ ws) {
  int b = blockIdx.y, row = blockIdx.x, t = threadIdx.x;
  const float* WT = ws + OFF_WTAB;
  float* DY = ws + OFF_DY;
  float* DX = ws + OFF_DX;
  float* CNT = ws + OFF_CNT;
  int y1 = 0, y2e = 0, x1 = 0, x2e = 0, cnt = 0;
  if (row < NQ) {
    int wi = wp[0], hi = hp[0];
    float wf = (float)wi, hfv = (float)hi;
    const float* bx = pb + (size_t)(b * NQ + row) * 4;
    float cx = bx[0], cy = bx[1], bw = bx[2], bh = bx[3];
    float bx1 = (cx - bw * 0.5f) * wf, by1 = (cy - bh * 0.5f) * hfv;
    float bx2 = (cx + bw * 0.5f) * wf, by2 = (cy + bh * 0.5f) * hfv;
    x1 = min(max((int)bx1, 0), wi);           /* trunc-toward-zero == astype(i32) */
    y1 = min(max((int)by1, 0), hi);
    int x2 = min(max((int)bx2, 0), wi);
    int y2 = min(max((int)by2, 0), hi);
    cnt = max(y2 - y1, 0) * max(x2 - x1, 0);
    x2e = max(x2, x1); y2e = max(y2, y1);
  }
  if (t < KPAD) {                             /* Dy: fold 1/2048 channel mean here */
    float v = 0.f;
    if (row < NQ && t < HF)
      v = (WT[y2e * HF + t] - WT[y1 * HF + t]) * (1.f / 2048.f);
    DY[((size_t)b * MPAD + row) * KPAD + t] = v;
  }
  if (row < NQ && t < HF)
    DX[((size_t)b * NQ + row) * HF + t] = WT[x2e * HF + t] - WT[x1 * HF + t];
  if (row < NQ && t == 0) CNT[b * NQ + row] = (float)cnt;
}

/* ---- WMMA GEMM: U[b] (304x48) = Dy[b] (304x44) @ F[b] (44x48) -------
 * One wave per 16x16 tile; K advanced 4 at a time via
 * V_WMMA_F32_16X16X4_F32 (wave32, layouts per ISA 7.12.2).             */
__global__ void k_gemm(float* __restrict__ ws) {
  const float* DY = ws + OFF_DY;
  const float* FP = ws + OFF_FP;
  float* U = ws + OFF_U;
  int b = blockIdx.y;
  int mt = blockIdx.x / 3, nt = blockIdx.x % 3;
  int lane = threadIdx.x;
  int half = lane >> 4, li = lane & 15;
  const float* A = DY + (size_t)b * MPAD * KPAD;
  const float* B = FP + (size_t)b * KPAD * NPAD;
  int arow = mt * 16 + li;
  int bcol = nt * 16 + li;
  v8f c = {};
  for (int kk = 0; kk < 11; ++kk) {
    int kb = kk * 4 + half * 2;               /* lanes 0-15: K0/K1, 16-31: K2/K3 */
    v2f a, bm;
    a.x = A[(size_t)arow * KPAD + kb];
    a.y = A[(size_t)arow * KPAD + kb + 1];
    bm.x = B[(size_t)kb * NPAD + bcol];
    bm.y = B[(size_t)(kb + 1) * NPAD + bcol];
    c = __builtin_amdgcn_wmma_f32_16x16x4_f32(false, a, false, bm,
                                              (short)0, c, false, false);
  }
  float* Ut = U + (size_t)b * MPAD * NPAD;
  int n = nt * 16 + li;
  #pragma unroll
  for (int v = 0; v < 8; ++v) {               /* VGPR v: M=v (lo half) / v+8 (hi) */
    int m = mt * 16 + v + half * 8;
    Ut[(size_t)m * NPAD + n] = c[v];
  }
}

/* ---- boxsum = U . Dx ; means with mask -------------------------------*/
__global__ void k_means(float* __restrict__ ws, const int* __restrict__ qidx) {
  int t = blockIdx.x * blockDim.x + threadIdx.x;
  if (t >= NB * NQ) return;
  int b = t / NQ, q = t % NQ;
  const float* Ur = ws + OFF_U + ((size_t)b * MPAD + q) * NPAD;
  const float* Dxr = ws + OFF_DX + ((size_t)b * NQ + q) * HF;
  float boxsum = 0.f;
  #pragma unroll 6
  for (int s = 0; s < HF; ++s) boxsum += Ur[s] * Dxr[s];
  float cnt = ws[OFF_CNT + b * NQ + q];
  bool matched = false;
  for (int i = 0; i < MM; ++i)
    if (qidx[b * MM + i] == q) matched = true;
  float m = (cnt > 0.f) ? boxsum / fmaxf(cnt, 1.f) : NEGV;
  ws[OFF_MEANS + b * NQ + q] = matched ? NEGV : m;
}

/* ---- loss tail --------------------------------------------------------*/
__device__ __forceinline__ float log_sigmoid(float x) {
  return (x >= 0.f) ? -log1pf(expf(-x)) : (x - log1pf(expf(x)));
}
__device__ __forceinline__ float log1m_sigmoid(float x) {
  return (x <= 0.f) ? -log1pf(expf(x)) : (-x - log1pf(expf(-x)));
}
__device__ float double_softmax_logp(const float* __restrict__ lp, int cls) {
  /* log_softmax(softmax(lp[0:91]))[cls] */
  float m1 = -1e30f;
  for (int j = 0; j < NCLS; ++j) m1 = fmaxf(m1, lp[j]);
  float S1 = 0.f;
  for (int j = 0; j < NCLS; ++j) S1 += expf(lp[j] - m1);
  float inv = 1.f / S1;
  float mp = inv;                              /* max prob = exp(0)/S1 */
  float S2 = 0.f;
  for (int j = 0; j < NCLS; ++j) S2 += expf(expf(lp[j] - m1) * inv - mp);
  float pc = expf(lp[cls] - m1) * inv;
  return pc - mp - logf(S2);
}

__global__ void k_loss(const float* __restrict__ logits, const float* __restrict__ pboxes,
                       const int* __restrict__ tlab, const float* __restrict__ tbox,
                       const int* __restrict__ qidx, const int* __restrict__ tidx,
                       const int* __restrict__ hp, const int* __restrict__ wp,
                       float* __restrict__ ws, float* __restrict__ out) {
  __shared__ float acc[8];   /* 0:ce_m 1:bce_m 2:l1sq 3:iou 4:ce_p 5:bce_t 6:rest_s 7:rest_n */
  __shared__ int top5[NTOPK];
  __shared__ unsigned char flag[NQ];
  int b = blockIdx.x, t = threadIdx.x;
  const float* means = ws + OFF_MEANS + b * NQ;
  for (int q = t; q < NQ; q += blockDim.x) flag[q] = 0;
  if (t < 8) acc[t] = 0.f;
  __syncthreads();
  if (t < MM) flag[qidx[b * MM + t]] = 1;
  __syncthreads();
  if (t == 0) {                                 /* top-5 of means */
    for (int k = 0; k < NTOPK; ++k) {
      float best = -3.0e38f; int bi = 0;
      for (int q = 0; q < NQ; ++q)
        if (!(flag[q] & 2) && means[q] > best) { best = means[q]; bi = q; }
      flag[bi] |= 2; top5[k] = bi;
    }
  }
  __syncthreads();

  float wf = (float)wp[0], hf = (float)hp[0];
  if (t < MM) {                                 /* matched-query terms */
    int i = t;
    int q = qidx[b * MM + i];
    int ti = tidx[b * MM + i];
    int tc = tlab[b * MM + ti];
    const float* lp = logits + (size_t)(b * NQ + q) * CCH;
    atomicAdd(&acc[0], -double_softmax_logp(lp, tc));
    atomicAdd(&acc[1], -fmaxf(log_sigmoid(lp[CCH - 1]), -100.f));
    const float* qb = pboxes + (size_t)(b * NQ + q) * 4;
    const float* tb = tbox + (size_t)(b * MM + ti) * 4;
    float tnb[4] = { tb[0] / wf, tb[1] / hf, tb[2] / wf, tb[3] / hf };
    float ss = 0.f;
    #pragma unroll
    for (int d = 0; d < 4; ++d) { float df = qb[d] - tnb[d]; ss += df * df; }
    atomicAdd(&acc[2], ss);
    float ax1 = qb[0] - qb[2] * .5f, ay1 = qb[1] - qb[3] * .5f;
    float ax2 = qb[0] + qb[2] * .5f, ay2 = qb[1] + qb[3] * .5f;
    float tx1 = tnb[0] - tnb[2] * .5f, ty1 = tnb[1] - tnb[3] * .5f;
    float tx2 = tnb[0] + tnb[2] * .5f, ty2 = tnb[1] + tnb[3] * .5f;
    float ix1 = fmaxf(ax1, tx1), iy1 = fmaxf(ay1, ty1);
    float ix2 = fminf(ax2, tx2), iy2 = fminf(ay2, ty2);
    float inter = fmaxf(ix2 - ix1, 0.f) * fmaxf(iy2 - iy1, 0.f);
    float ua = (ax2 - ax1) * (ay2 - ay1) + (tx2 - tx1) * (ty2 - ty1) - inter + 1e-9f;
    atomicAdd(&acc[3], 1.f - inter / ua);
  }
  if (t >= 32 && t < 32 + NTOPK) {              /* pseudo-label terms */
    int q = top5[t - 32];
    const float* lp = logits + (size_t)(b * NQ + q) * CCH;
    atomicAdd(&acc[4], -double_softmax_logp(lp, NCLS - 1));
    atomicAdd(&acc[5], -fmaxf(log_sigmoid(lp[CCH - 1]), -100.f));
  }
  {                                             /* rest-query BCE */
    float s = 0.f; float n = 0.f;
    for (int q = t; q < NQ; q += blockDim.x) {
      if (flag[q] == 0) {
        float lobj = logits[(size_t)(b * NQ + q) * CCH + CCH - 1];
        s += -fmaxf(log1m_sigmoid(lobj), -100.f);
        n += 1.f;
      }
    }
    if (s != 0.f || n != 0.f) { atomicAdd(&acc[6], s); atomicAdd(&acc[7], n); }
  }
  __syncthreads();
  if (t == 0) {
    float ce_m = acc[0] / (float)MM, bce_m = acc[1] / (float)MM;
    float ce_p = acc[4] / (float)NTOPK, bce_t = acc[5] / (float)NTOPK;
    float rest = acc[6] / fmaxf(acc[7], 1.f);
    float l1 = sqrtf(acc[2]);
    float total = 2.f * (ce_m + ce_p) + 2.f * (bce_m + rest + bce_t)
                + 2.f * acc[3] + 5.f * l1;
    atomicAdd(out, total);
  }
}

extern "C" void kernel_launch(void* const* d_in, const int* in_sizes, int n_in,
                              void* d_out, int out_size, void* d_ws, size_t ws_size,
                              hipStream_t stream) {
  const float* img    = (const float*)d_in[0];
  const float* logits = (const float*)d_in[1];
  const float* pboxes = (const float*)d_in[2];
  const int*   tlab   = (const int*)d_in[3];
  const float* tbox   = (const float*)d_in[4];
  const int*   qidx   = (const int*)d_in[5];
  const int*   tidx   = (const int*)d_in[6];
  const int*   hp     = (const int*)d_in[7];
  const int*   wp     = (const int*)d_in[8];
  float* ws  = (float*)d_ws;
  float* out = (float*)d_out;

  k_init   <<<(NB * PIX + 255) / 256, 256, 0, stream>>>(ws, out);
  k_weights<<<1, 64, 0, stream>>>(ws);
  k_chansum<<<dim3(16, NB), 256, 0, stream>>>(img, ws);
  k_fpad   <<<NB, 256, 0, stream>>>(ws);
  k_boxvec <<<dim3(MPAD, NB), 48, 0, stream>>>(pboxes, hp, wp, ws);
  k_gemm   <<<dim3(19 * 3, NB), 32, 0, stream>>>(ws);
  k_means  <<<(NB * NQ + 127) / 128, 128, 0, stream>>>(ws, qidx);
  k_loss   <<<NB, 256, 0, stream>>>(logits, pboxes, tlab, tbox, qidx, tidx,
                                    hp, wp, ws, out);
}